// DLTSolver_48155173323001
// MI455X (gfx1250) — compile-verified
//
#include <hip/hip_runtime.h>

// DLT homography solver, MI455X (gfx1250).
//
// Roofline: B=1M elements, 32B in + 36B out = 68 MB => ~2.9us at 23.3 TB/s.
// The 8x8 solve collapses analytically (6 constant rows) to a 2x2 Cramer
// solve + back-substitution (~30 FLOPs/elem) => purely bandwidth bound,
// AI ~ 0.45 FLOP/byte, far below any WMMA crossover. The CDNA5 features
// that matter are the async LDS copy engine (ASYNCcnt path), bank-conflict-
// free LDS staging, and non-temporal coalesced streamout (don't pollute L2).
//
// All offsets fit in u32 (in = 32MB, out = 36MB), so index math is 32-bit.

#define TPB 256

__global__ __launch_bounds__(TPB) void dlt_solver_kernel(
    const float* __restrict__ in, float* __restrict__ out, unsigned int B) {
  __shared__ __align__(16) float lds_in[TPB * 8];   // 8 KB staged input
  __shared__ __align__(16) float lds_out[TPB * 9];  // 9 KB staged output

  const unsigned int tid = threadIdx.x;
  const unsigned int idx = blockIdx.x * TPB + tid;  // < 2^20, fits u32

  // ---- Stage 1: async copy this thread's 32B into LDS (gfx1250 path). ----
  if (idx < B) {
    // Low 32 bits of a generic LDS pointer == wave-relative LDS byte address
    // (flat LDS aperture truncation, ISA 10.2).
    unsigned int l0 = (unsigned int)(uintptr_t)(&lds_in[tid * 8]);
    // 64-bit base (SGPR) + 32-bit byte offset (VGPR): one v_add_nc_u64 max.
    unsigned long long g0 =
        (unsigned long long)(uintptr_t)in + (unsigned long long)(idx * 32u);
    asm volatile(
        "global_load_async_to_lds_b128 %0, %1, off\n\t"
        "global_load_async_to_lds_b128 %0, %1, off offset:16"
        :
        : "v"(l0), "v"(g0)
        : "memory");
  }
  // Per-wave ASYNCcnt: each thread reads back only its own bytes, so a
  // wave-scope wait suffices (no workgroup barrier needed before compute).
  asm volatile("s_wait_asynccnt 0" ::: "memory");

  // ---- Stage 2: closed-form solve. ----
  if (idx < B) {
    const float4 q0 = *reinterpret_cast<const float4*>(&lds_in[tid * 8]);
    const float4 q1 = *reinterpret_cast<const float4*>(&lds_in[tid * 8 + 4]);

    // pred = shift + ref, ref = [0,0,512,0,0,512,512,512]
    const float p0 = q0.x;
    const float p1 = q0.y;
    const float p2 = q0.z + 512.0f;
    const float p3 = q0.w;
    const float p4 = q1.x;
    const float p5 = q1.y + 512.0f;
    const float p6 = q1.z + 512.0f;
    const float p7 = q1.w + 512.0f;

    const float is = 1.0f / 512.0f;

    // Rows 6,7 of the DLT system reduced onto (h6,h7):
    //   (p7-p3) h6 + (p2-p6) h7 = (p2*(p3-p1) - p7)/512
    //   (p7-p5) h6 + (p4-p6) h7 = (p6 - p5^2 + p0*p4)/512
    const float A00 = p7 - p3, A01 = p2 - p6;
    const float A10 = p7 - p5, A11 = p4 - p6;
    const float r0 = (p2 * (p3 - p1) - p7) * is;
    const float r1 = (p6 - p5 * p5 + p0 * p4) * is;

    const float det = A00 * A11 - A01 * A10;
    const float invdet = 1.0f / det;
    const float h6 = (r0 * A11 - A01 * r1) * invdet;
    const float h7 = (A00 * r1 - A10 * r0) * invdet;

    // Back-substitution (rows 0,1,3,4 then rows 2,5).
    const float h3 = -p1 * is - h7;
    const float h5 = p0 * is - h7;
    const float h2 = -p2 * is - h6;
    const float h4 = p5 * is - h6;
    const float h1 = (p1 - p0) * is - p3 + h7;
    const float h0 = (p2 - p5) * is - p4 + h6;

    // Stride-9 staging: gcd(9,64)==1 => bank-conflict free across a wave.
    float* o = &lds_out[tid * 9];
    o[0] = h0; o[1] = h1; o[2] = h2;
    o[3] = h3; o[4] = h4; o[5] = h5;
    o[6] = h6; o[7] = h7; o[8] = 1.0f;
  }

  __syncthreads();

  // ---- Stage 3: coalesced, non-temporal streamout of 2304 dwords/block. ----
  const unsigned int obase = blockIdx.x * (TPB * 9);  // <= 9*2^20, fits u32
  const unsigned int total = B * 9u;
#pragma unroll
  for (int k = 0; k < 9; ++k) {
    const unsigned int o = obase + (unsigned int)k * TPB + tid;
    if (o < total) {
      __builtin_nontemporal_store(lds_out[k * TPB + tid], out + o);
    }
  }
}

extern "C" void kernel_launch(void* const* d_in, const int* in_sizes, int n_in,
                              void* d_out, int out_size, void* d_ws, size_t ws_size,
                              hipStream_t stream) {
  const float* in = (const float*)d_in[0];
  float* out = (float*)d_out;
  const unsigned int B = (unsigned int)(in_sizes[0] / 8);
  const int grid = (int)((B + TPB - 1) / TPB);
  dlt_solver_kernel<<<grid, TPB, 0, stream>>>(in, out, B);
}